// MoGIN14_48266842472589
// MI455X (gfx1250) — compile-verified
//
#include <hip/hip_runtime.h>
#include <stdint.h>

// ---------------------------------------------------------------------------
// Model constants (from the reference)
// ---------------------------------------------------------------------------
#define K_NODES   10000
#define K_EDGES   320000
#define K_GRAPHS  32
#define K_DNODE   128
#define K_DEDGE   64
#define K_HNODE   256
#define K_HEDGE   128
#define K_HROUTER 64
#define K_EXPERTS 8
#define K_DMSG    192
#define K_EPS     1e-5f
#define K_RFILT   0.05f
#define K_LMEAN   2.7554f
#define K_LSTD    1.1664f
#define K_LBA     0.1f
#define K_LBB     0.02f
#define K_LBT     0.01f

// epilogue flags for the fused WMMA GEMM
#define GF_LN    1
#define GF_SILU  2
#define GF_ADD   4
#define GF_ACCUM 8

typedef __attribute__((ext_vector_type(2))) float v2f;
typedef __attribute__((ext_vector_type(8))) float v8f;

__device__ __forceinline__ float silu_f(float x) { return x / (1.0f + expf(-x)); }

// ---------------------------------------------------------------------------
// utility kernels
// ---------------------------------------------------------------------------
__global__ void zero_kernel(float* __restrict__ p, long long n) {
  long long i = (long long)blockIdx.x * blockDim.x + threadIdx.x;
  if (i < n) p[i] = 0.0f;
}

__global__ void nel_kernel(const float* __restrict__ pos,
                           const int* __restrict__ src, const int* __restrict__ dst,
                           float* __restrict__ nel) {
  int e = blockIdx.x * blockDim.x + threadIdx.x;
  if (e >= K_EDGES) return;
  int s = src[e], d = dst[e];
  float dx = pos[3 * s + 0] - pos[3 * d + 0];
  float dy = pos[3 * s + 1] - pos[3 * d + 1];
  float dz = pos[3 * s + 2] - pos[3 * d + 2];
  nel[e] = (sqrtf(dx * dx + dy * dy + dz * dz) - K_LMEAN) / K_LSTD;
}

__global__ void gather_h_kernel(const int* __restrict__ atom_type,
                                const float* __restrict__ emb,
                                float* __restrict__ h) {
  int i = blockIdx.x * blockDim.x + threadIdx.x;
  if (i >= K_NODES * K_DNODE) return;
  int n = i >> 7, c = i & 127;
  h[i] = emb[(size_t)atom_type[n] * K_DNODE + c];
}

// ---------------------------------------------------------------------------
// per-edge router (1->64 LN silu ->8 softmax) + edge MLP (1->128 LN silu ->64)
// also accumulates load-balance statistics: lb[0..7] = sum_e route, lb[8] = sum ||route||^2
// ---------------------------------------------------------------------------
__global__ __launch_bounds__(256)
void edge_router_kernel(const float* __restrict__ nel,
                        const float* __restrict__ rW0, const float* __restrict__ rb0,
                        const float* __restrict__ rg0, const float* __restrict__ rbe0,
                        const float* __restrict__ rW1, const float* __restrict__ rb1,
                        const float* __restrict__ eW0, const float* __restrict__ eb0,
                        const float* __restrict__ eg0, const float* __restrict__ ebe0,
                        const float* __restrict__ eW1, const float* __restrict__ eb1,
                        float* __restrict__ route, float* __restrict__ ea,
                        float* __restrict__ lb_acc) {
  __shared__ float sRW1[K_HROUTER * K_EXPERTS];   // 64x8
  __shared__ float sEW1[K_HEDGE * K_DEDGE];       // 128x64
  __shared__ float part[9];
  const int tid = threadIdx.x;
  for (int i = tid; i < K_HROUTER * K_EXPERTS; i += 256) sRW1[i] = rW1[i];
  for (int i = tid; i < K_HEDGE * K_DEDGE; i += 256) sEW1[i] = eW1[i];
  if (tid < 9) part[tid] = 0.0f;
  __syncthreads();

  const int e = blockIdx.x * 256 + tid;
  if (e < K_EDGES) {
    const float t = nel[e];

    // ---- router ----
    float hr[K_HROUTER];
    float mu = 0.0f;
    for (int i = 0; i < K_HROUTER; ++i) { hr[i] = t * rW0[i] + rb0[i]; mu += hr[i]; }
    mu *= (1.0f / K_HROUTER);
    float var = 0.0f;
    for (int i = 0; i < K_HROUTER; ++i) { float d = hr[i] - mu; var += d * d; }
    const float rs = rsqrtf(var * (1.0f / K_HROUTER) + K_EPS);
    for (int i = 0; i < K_HROUTER; ++i) {
      float x = (hr[i] - mu) * rs * rg0[i] + rbe0[i];
      hr[i] = silu_f(x);
    }
    float ro[K_EXPERTS];
    float mx = -1e30f;
    for (int j = 0; j < K_EXPERTS; ++j) {
      float s = rb1[j];
      for (int i = 0; i < K_HROUTER; ++i) s += hr[i] * sRW1[i * K_EXPERTS + j];
      ro[j] = s;
      mx = fmaxf(mx, s);
    }
    float se = 0.0f;
    for (int j = 0; j < K_EXPERTS; ++j) { ro[j] = expf(ro[j] - mx); se += ro[j]; }
    const float inv = 1.0f / se;
    float ssq = 0.0f;
    for (int j = 0; j < K_EXPERTS; ++j) {
      float p = ro[j] * inv;
      route[(size_t)e * K_EXPERTS + j] = p;
      ssq += p * p;
      atomicAdd(&part[j], p);
    }
    atomicAdd(&part[8], ssq);

    // ---- edge MLP ----
    float he[K_HEDGE];
    mu = 0.0f;
    for (int i = 0; i < K_HEDGE; ++i) { he[i] = t * eW0[i] + eb0[i]; mu += he[i]; }
    mu *= (1.0f / K_HEDGE);
    var = 0.0f;
    for (int i = 0; i < K_HEDGE; ++i) { float d = he[i] - mu; var += d * d; }
    const float rs2 = rsqrtf(var * (1.0f / K_HEDGE) + K_EPS);
    for (int i = 0; i < K_HEDGE; ++i) {
      float x = (he[i] - mu) * rs2 * eg0[i] + ebe0[i];
      he[i] = silu_f(x);
    }
    for (int j = 0; j < K_DEDGE; ++j) {
      float s = eb1[j];
      for (int i = 0; i < K_HEDGE; ++i) s += he[i] * sEW1[i * K_DEDGE + j];
      ea[(size_t)e * K_DEDGE + j] = s;
    }
  }
  __syncthreads();
  if (tid < 9) atomicAdd(&lb_acc[tid], part[tid]);
}

__global__ void lb_finish_kernel(const float* __restrict__ lb_acc, float* __restrict__ total) {
  if (blockIdx.x != 0 || threadIdx.x != 0) return;
  const float frac = 1.0f / K_EXPERTS;
  float s = 0.0f;
  for (int j = 0; j < K_EXPERTS; ++j) {
    float m = lb_acc[j] * (1.0f / K_EDGES);
    s += m * m;
  }
  float ebl = (s - frac) * (1.0f / (1.0f - frac));
  float unc = 1.0f - lb_acc[8] * (1.0f / K_EDGES);
  float t = (K_LBA + K_LBB) * K_LBT;
  float unf = K_LBA * ebl + K_LBB * unc;
  float lbl = (fmaxf(unf, t) - t) * ((K_LBA + K_LBB) / (K_LBA + K_LBB - t));
  atomicAdd(total, lbl);
}

// ---------------------------------------------------------------------------
// fused message passing: one pass accumulates shared (w=1) + 8 expert-weighted
// scatter-adds of m = concat(h[src], ea) into acc[9][K_NODES][192] (L2 resident).
// one wave32 per edge; lane-uniform route weight so no divergence inside a wave.
// ---------------------------------------------------------------------------
__global__ __launch_bounds__(256)
void prop_kernel(const float* __restrict__ h, const float* __restrict__ ea,
                 const float* __restrict__ route,
                 const int* __restrict__ src, const int* __restrict__ dst,
                 float* __restrict__ acc) {
  const int e = blockIdx.x * 8 + (threadIdx.x >> 5);
  if (e >= K_EDGES) return;
  const int lane = threadIdx.x & 31;
  const int s = src[e];
  const int d = dst[e];
  float m[6];
#pragma unroll
  for (int i = 0; i < 4; ++i) m[i] = h[(size_t)s * K_DNODE + lane + 32 * i];
  m[4] = ea[(size_t)e * K_DEDGE + lane];
  m[5] = ea[(size_t)e * K_DEDGE + lane + 32];
#pragma unroll
  for (int j = 0; j < 9; ++j) {
    float w = 1.0f;
    if (j > 0) {
      w = route[(size_t)e * K_EXPERTS + (j - 1)];
      if (!(w > K_RFILT)) continue;  // route filter (uniform across the wave)
    }
    float* base = acc + ((size_t)j * K_NODES + d) * K_DMSG;
#pragma unroll
    for (int i = 0; i < 6; ++i) atomicAdd(base + lane + 32 * i, w * m[i]);
  }
}

// ---------------------------------------------------------------------------
// Fused WMMA GEMM: Z = act( LN?( X[16xK] @ W[KxN] + b (+ addend) ) ), one
// workgroup (4 waves) = 16 rows x full N, so the row LayerNorm reduces in LDS.
// Uses V_WMMA_F32_16X16X4_F32 (fp32 = reference precision).
// X tile is staged into LDS with GLOBAL_LOAD_ASYNC_TO_LDS_B128 (ASYNCcnt path):
// M = 10000 = 625*16 exactly and every X base/row stride is 16B-aligned, so the
// whole 16xK tile is copied with uniform, in-bounds b128 async transfers.
// A frag: lane(half=l>>4, r=l&15) -> A[m0+r][k0+2h .. +1]
// B frag: -> B[k0+2h .. +1][n0+r] ;  D vgpr v -> C[m0+v+8h][n0+r]
// ---------------------------------------------------------------------------
template <int K, int N>
__global__ __launch_bounds__(128)
void gemm16_fused(const float* __restrict__ X, int xStride,
                  const float* __restrict__ W, const float* __restrict__ bias,
                  const float* __restrict__ addend,
                  const float* __restrict__ g, const float* __restrict__ be,
                  float* __restrict__ out, int flags) {
  __shared__ float xs[16][K];
  __shared__ float zs[16][N];
  __shared__ float rowMu[16];
  __shared__ float rowRs[16];

  const int tid = threadIdx.x;
  const int wave = tid >> 5;
  const int lane = tid & 31;
  const int half = lane >> 4;
  const int r = lane & 15;
  const int m0 = blockIdx.x * 16;

  static_assert(K_NODES % 16 == 0, "exact M tiling required for async staging");
  {
    // 16*K floats = 16*K/4 float4 transfers; K/32 per thread, no divergence.
    constexpr int F4 = 16 * K / 4;
    constexpr int ITER = F4 / 128;
    const unsigned ldsBase = (unsigned)(uintptr_t)(&xs[0][0]);  // addr[31:0] = LDS offset
#pragma unroll
    for (int it = 0; it < ITER; ++it) {
      const int f = tid + it * 128;
      const int row = f / (K / 4);
      const int c4 = f - row * (K / 4);
      const float* gp = X + (size_t)(m0 + row) * xStride + c4 * 4;
      const unsigned lds = ldsBase + (unsigned)(f * 16);
      asm volatile("global_load_async_to_lds_b128 %0, %1, off"
                   :: "v"(lds), "v"(gp)
                   : "memory");
    }
    asm volatile("s_wait_asynccnt 0x0" ::: "memory");
  }
  __syncthreads();

  constexpr int NT = N / 16;
  constexpr int TPW = NT / 4;  // N in {128,192,256}: divides evenly -> no divergence
#pragma unroll
  for (int t = 0; t < TPW; ++t) {
    const int n0 = (wave * TPW + t) * 16;
    v8f acc = {};
    for (int k0 = 0; k0 < K; k0 += 4) {
      v2f a, b;
      a[0] = xs[r][k0 + 2 * half + 0];
      a[1] = xs[r][k0 + 2 * half + 1];
      b[0] = W[(size_t)(k0 + 2 * half + 0) * N + n0 + r];
      b[1] = W[(size_t)(k0 + 2 * half + 1) * N + n0 + r];
      acc = __builtin_amdgcn_wmma_f32_16x16x4_f32(false, a, false, b, (short)0, acc,
                                                  false, false);
    }
    const float bv = bias[n0 + r];
#pragma unroll
    for (int v = 0; v < 8; ++v) {
      const int row = v + 8 * half;
      float z = acc[v] + bv;
      if (flags & GF_ADD) {
        int gm = m0 + row;
        if (gm < K_NODES) z += addend[(size_t)gm * N + n0 + r];
      }
      zs[row][n0 + r] = z;
    }
  }
  __syncthreads();

  if (flags & GF_LN) {
    if (tid < 16) {
      float s = 0.0f;
      for (int c = 0; c < N; ++c) s += zs[tid][c];
      float mu = s * (1.0f / N);
      float v = 0.0f;
      for (int c = 0; c < N; ++c) { float d = zs[tid][c] - mu; v += d * d; }
      rowMu[tid] = mu;
      rowRs[tid] = rsqrtf(v * (1.0f / N) + K_EPS);
    }
    __syncthreads();
  }

  for (int i = tid; i < 16 * N; i += 128) {
    int row = i / N, col = i - row * N;
    int gm = m0 + row;
    if (gm >= K_NODES) continue;
    float z = zs[row][col];
    if (flags & GF_LN) z = (z - rowMu[row]) * rowRs[row] * g[col] + be[col];
    if (flags & GF_SILU) z = silu_f(z);
    float* o = out + (size_t)gm * N + col;
    if (flags & GF_ACCUM) *o += z;
    else *o = z;
  }
}

// ---------------------------------------------------------------------------
// GraphNorm (per-graph, per-channel) + residual
// ---------------------------------------------------------------------------
__global__ void gnorm_pass1(const float* __restrict__ x, const int* __restrict__ batch,
                            float* __restrict__ gsum, float* __restrict__ gcnt) {
  int i = blockIdx.x * blockDim.x + threadIdx.x;
  if (i >= K_NODES * K_DNODE) return;
  int n = i >> 7, c = i & 127;
  int b = batch[n];
  atomicAdd(&gsum[b * K_DNODE + c], x[i]);
  if (c == 0) atomicAdd(&gcnt[b], 1.0f);
}

__global__ void gnorm_pass2(const float* __restrict__ x, const int* __restrict__ batch,
                            const float* __restrict__ ms, const float* __restrict__ gsum,
                            const float* __restrict__ gcnt, float* __restrict__ gsq) {
  int i = blockIdx.x * blockDim.x + threadIdx.x;
  if (i >= K_NODES * K_DNODE) return;
  int n = i >> 7, c = i & 127;
  int b = batch[n];
  float mean = gsum[b * K_DNODE + c] / gcnt[b];
  float o = x[i] - ms[c] * mean;
  atomicAdd(&gsq[b * K_DNODE + c], o * o);
}

__global__ void gnorm_pass3(const float* __restrict__ x, const int* __restrict__ batch,
                            const float* __restrict__ ms, const float* __restrict__ wv,
                            const float* __restrict__ bv, const float* __restrict__ gsum,
                            const float* __restrict__ gcnt, const float* __restrict__ gsq,
                            const float* __restrict__ h_in, const float* __restrict__ alpha,
                            int li, int final_act, float* __restrict__ h_out) {
  int i = blockIdx.x * blockDim.x + threadIdx.x;
  if (i >= K_NODES * K_DNODE) return;
  int n = i >> 7, c = i & 127;
  int b = batch[n];
  float cnt = gcnt[b];
  float mean = gsum[b * K_DNODE + c] / cnt;
  float o = x[i] - ms[c] * mean;
  float var = gsq[b * K_DNODE + c] / cnt;
  float y = wv[c] * o * rsqrtf(var + K_EPS) + bv[c];
  if (final_act) y = silu_f(y);
  h_out[i] = h_in[i] + alpha[li] * y;
}

// ---------------------------------------------------------------------------
// host driver
// ---------------------------------------------------------------------------
extern "C" void kernel_launch(void* const* d_in, const int* in_sizes, int n_in,
                              void* d_out, int out_size, void* d_ws, size_t ws_size,
                              hipStream_t stream) {
  (void)in_sizes; (void)n_in; (void)out_size; (void)ws_size;

  // Input order: top-level setup_inputs() insertion order; `params` expanded
  // in JAX pytree leaf order (dict keys ASCII-sorted at every level).
  int k = 0;
  const float* pos      = (const float*)d_in[k++];
  const float* alpha    = (const float*)d_in[k++];   // params.alpha
  const float* atom_emb = (const float*)d_in[k++];   // params.atom_emb

  struct Layer {
    const float *eW0, *eW1, *eb0, *eb1, *ebe0, *ebf;
    const float *edW0, *edW1, *edb0, *edb1, *edbe0, *edg0;
    const float *eg0, *egf;
    const float *gn_b, *gn_ms, *gn_w;
    const float *pW, *pb;
    const float *rW0, *rW1, *rb0, *rb1, *rbe0, *rg0;
    const float *sW0, *sW1, *sb0, *sb1, *sbe0, *sbf, *sg0, *sgf;
  } L[3];

  for (int i = 0; i < 3; ++i) {
    const bool last = (i == 2);
    L[i].eW0  = (const float*)d_in[k++];
    L[i].eW1  = (const float*)d_in[k++];
    L[i].eb0  = (const float*)d_in[k++];
    L[i].eb1  = (const float*)d_in[k++];
    L[i].ebe0 = (const float*)d_in[k++];
    L[i].ebf  = last ? nullptr : (const float*)d_in[k++];
    // edge sub-dict: W0, W1, b0, b1, be0, g0
    L[i].edW0  = (const float*)d_in[k++];
    L[i].edW1  = (const float*)d_in[k++];
    L[i].edb0  = (const float*)d_in[k++];
    L[i].edb1  = (const float*)d_in[k++];
    L[i].edbe0 = (const float*)d_in[k++];
    L[i].edg0  = (const float*)d_in[k++];
    L[i].eg0 = (const float*)d_in[k++];
    L[i].egf = last ? nullptr : (const float*)d_in[k++];
    L[i].gn_b  = (const float*)d_in[k++];
    L[i].gn_ms = (const float*)d_in[k++];
    L[i].gn_w  = (const float*)d_in[k++];
    L[i].pW = (const float*)d_in[k++];
    L[i].pb = (const float*)d_in[k++];
    // router sub-dict: W0, W1, b0, b1, be0, g0
    L[i].rW0  = (const float*)d_in[k++];
    L[i].rW1  = (const float*)d_in[k++];
    L[i].rb0  = (const float*)d_in[k++];
    L[i].rb1  = (const float*)d_in[k++];
    L[i].rbe0 = (const float*)d_in[k++];
    L[i].rg0  = (const float*)d_in[k++];
    // shared sub-dict: W0, W1, b0, b1, be0, [bf], g0, [gf]
    L[i].sW0  = (const float*)d_in[k++];
    L[i].sW1  = (const float*)d_in[k++];
    L[i].sb0  = (const float*)d_in[k++];
    L[i].sb1  = (const float*)d_in[k++];
    L[i].sbe0 = (const float*)d_in[k++];
    L[i].sbf  = last ? nullptr : (const float*)d_in[k++];
    L[i].sg0  = (const float*)d_in[k++];
    L[i].sgf  = last ? nullptr : (const float*)d_in[k++];
  }
  const int* atom_type  = (const int*)d_in[k++];
  const int* batch      = (const int*)d_in[k++];
  const int* edge_index = (const int*)d_in[k++];
  const int* src = edge_index;
  const int* dst = edge_index + K_EDGES;

  // workspace layout (floats, contiguous; every block is a multiple of 4
  // floats so all base pointers stay 16B-aligned for the b128 async copies)
  float* w = (float*)d_ws;
  auto take = [&](size_t n) { float* p = w; w += n; return p; };
  float* nel   = take(K_EDGES);
  float* route = take((size_t)K_EDGES * K_EXPERTS);
  float* ea    = take((size_t)K_EDGES * K_DEDGE);
  float* acc   = take((size_t)9 * K_NODES * K_DMSG);
  float* hA    = take((size_t)K_NODES * K_DNODE);
  float* hB    = take((size_t)K_NODES * K_DNODE);
  float* outb  = take((size_t)K_NODES * K_DNODE);
  float* z192  = take((size_t)K_NODES * K_DMSG);
  float* t256  = take((size_t)K_NODES * K_HNODE);
  float* gsum  = take(K_GRAPHS * K_DNODE);
  float* gsq   = take(K_GRAPHS * K_DNODE);
  float* gcnt  = take(K_GRAPHS);
  float* lb    = take(16);

  float* out_h     = (float*)d_out;
  float* out_total = out_h + (size_t)K_NODES * K_DNODE;

  zero_kernel<<<1, 32, 0, stream>>>(out_total, 1);
  nel_kernel<<<K_EDGES / 256, 256, 0, stream>>>(pos, src, dst, nel);
  gather_h_kernel<<<(K_NODES * K_DNODE) / 256, 256, 0, stream>>>(atom_type, atom_emb, hA);

  float* cur = hA;
  float* nxt = hB;
  for (int i = 0; i < 3; ++i) {
    const Layer& P = L[i];
    const bool last = (i == 2);
    const int fa = last ? 0 : 1;          // final_act
    const int flags2 = last ? 0 : (GF_LN | GF_SILU);

    // edge features, routing probabilities, load-balance stats
    zero_kernel<<<1, 32, 0, stream>>>(lb, 16);
    edge_router_kernel<<<K_EDGES / 256, 256, 0, stream>>>(
        nel, P.rW0, P.rb0, P.rg0, P.rbe0, P.rW1, P.rb1,
        P.edW0, P.edb0, P.edg0, P.edbe0, P.edW1, P.edb1, route, ea, lb);
    lb_finish_kernel<<<1, 32, 0, stream>>>(lb, out_total);

    // fused 9-way message passing into L2-resident accumulators
    const long long accN = (long long)9 * K_NODES * K_DMSG;
    zero_kernel<<<(int)((accN + 255) / 256), 256, 0, stream>>>(acc, accN);
    prop_kernel<<<K_EDGES / 8, 256, 0, stream>>>(cur, ea, route, src, dst, acc);

    // shared expert: acc[0] -> 256 (LN,silu) -> 128 (+LN,+silu unless last)
    gemm16_fused<K_DMSG, K_HNODE><<<K_NODES / 16, 128, 0, stream>>>(
        acc, K_DMSG, P.sW0, P.sb0, nullptr, P.sg0, P.sbe0, t256, GF_LN | GF_SILU);
    gemm16_fused<K_HNODE, K_DNODE><<<K_NODES / 16, 128, 0, stream>>>(
        t256, K_HNODE, P.sW1, P.sb1, nullptr, P.sgf, P.sbf, outb, flags2);

    // experts: z = acc[1+e] + h@pW+pb ; -> 256 (LN,silu) ; -> 128 (+LN,+silu) += out
    for (int e = 0; e < K_EXPERTS; ++e) {
      const float* accE = acc + (size_t)(1 + e) * K_NODES * K_DMSG;
      gemm16_fused<K_DNODE, K_DMSG><<<K_NODES / 16, 128, 0, stream>>>(
          cur, K_DNODE, P.pW + (size_t)e * K_DNODE * K_DMSG, P.pb + e * K_DMSG,
          accE, nullptr, nullptr, z192, GF_ADD);
      gemm16_fused<K_DMSG, K_HNODE><<<K_NODES / 16, 128, 0, stream>>>(
          z192, K_DMSG, P.eW0 + (size_t)e * K_DMSG * K_HNODE, P.eb0 + e * K_HNODE,
          nullptr, P.eg0 + e * K_HNODE, P.ebe0 + e * K_HNODE, t256, GF_LN | GF_SILU);
      gemm16_fused<K_HNODE, K_DNODE><<<K_NODES / 16, 128, 0, stream>>>(
          t256, K_HNODE, P.eW1 + (size_t)e * K_HNODE * K_DNODE, P.eb1 + e * K_DNODE,
          nullptr, last ? nullptr : (P.egf + e * K_DNODE),
          last ? nullptr : (P.ebf + e * K_DNODE), outb, flags2 | GF_ACCUM);
    }

    // GraphNorm + residual
    const long long gN = (long long)K_GRAPHS * K_DNODE * 2 + K_GRAPHS;
    zero_kernel<<<(int)((gN + 255) / 256), 256, 0, stream>>>(gsum, gN);
    gnorm_pass1<<<(K_NODES * K_DNODE) / 256, 256, 0, stream>>>(outb, batch, gsum, gcnt);
    gnorm_pass2<<<(K_NODES * K_DNODE) / 256, 256, 0, stream>>>(outb, batch, P.gn_ms,
                                                               gsum, gcnt, gsq);
    float* hdst = last ? out_h : nxt;
    gnorm_pass3<<<(K_NODES * K_DNODE) / 256, 256, 0, stream>>>(
        outb, batch, P.gn_ms, P.gn_w, P.gn_b, gsum, gcnt, gsq, cur, alpha, i, fa, hdst);
    if (!last) { float* ts = cur; cur = nxt; nxt = ts; }
  }
}